// Ortho_Trans_37984690765992
// MI455X (gfx1250) — compile-verified
//
#include <hip/hip_runtime.h>
#include <hip/hip_bf16.h>

typedef float v2f __attribute__((ext_vector_type(2)));
typedef float v8f __attribute__((ext_vector_type(8)));

#define GROUPS 64
#define C 32
#define DDIM 18432
#define KC 64
#define SPLIT 16                  // K-split blocks per group (18432/16 = 1152 = 18*KC)
#define NCHUNK ((DDIM / SPLIT) / KC)
#define NBUF 4                    // async pipeline depth (3 chunks in flight)
#define T_ITERS 5
#define EPS_VAL 1e-5f
#define ZPAD 68   // 272B row stride: 16B-aligned rows for async B128, conflict-free
#define SPAD (C + 2)

static_assert(NCHUNK >= NBUF, "pipeline deeper than chunk count");

__device__ __forceinline__ v8f wmma4(v2f a, v2f b, v8f c) {
    return __builtin_amdgcn_wmma_f32_16x16x4_f32(false, a, false, b,
                                                 (short)0, c, false, false);
}

// 16x16 tile of X(32x32) @ Y(32x32), LDS row-major, SPAD stride.
__device__ __forceinline__ v8f mm_tile(const float (*X)[SPAD], const float (*Y)[SPAD],
                                       int ra0, int rb0, int lane) {
    v8f acc = {};
    const int r  = lane & 15;
    const int hi = (lane >= 16) ? 2 : 0;
#pragma unroll
    for (int kk = 0; kk < C; kk += 4) {
        v2f a = *(const v2f*)&X[ra0 + r][kk + hi];
        v2f b;
        b.x = Y[kk + hi + 0][rb0 + r];
        b.y = Y[kk + hi + 1][rb0 + r];
        acc = wmma4(a, b, acc);
    }
    return acc;
}

__device__ __forceinline__ void store_tile(float (*Z)[SPAD], const v8f& acc,
                                           int ra0, int rb0, int lane) {
    const int r  = lane & 15;
    const int mo = (lane >= 16) ? 8 : 0;
#pragma unroll
    for (int v = 0; v < 8; ++v) Z[ra0 + v + mo][rb0 + r] = acc[v];
}

// Issue one 32xKC f32 chunk as async global->LDS B128 copies (4 per thread).
__device__ __forceinline__ void issue_chunk_async(const float* __restrict__ Zg, int k0,
                                                  float (*buf)[ZPAD], int tid) {
#pragma unroll
    for (int j = 0; j < 4; ++j) {
        int le  = tid + j * 128;          // float4 linear index, 0..511
        int row = le >> 4;                // 16 float4 per row
        int c4  = (le & 15) << 2;
        unsigned      lo = (unsigned)(size_t)&buf[row][c4];   // LDS byte offset
        const float*  gp = Zg + (size_t)row * DDIM + k0 + c4; // 16B aligned
        asm volatile("global_load_async_to_lds_b128 %0, %1, off"
                     :: "v"(lo), "v"(gp) : "memory");
    }
}

__device__ __forceinline__ void wait_async_le12() {
#if __has_builtin(__builtin_amdgcn_s_wait_asynccnt)
    __builtin_amdgcn_s_wait_asynccnt(12);
#else
    asm volatile("s_wait_asynccnt 0xc" ::: "memory");
#endif
}
__device__ __forceinline__ void wait_async_0() {
#if __has_builtin(__builtin_amdgcn_s_wait_asynccnt)
    __builtin_amdgcn_s_wait_asynccnt(0);
#else
    asm volatile("s_wait_asynccnt 0x0" ::: "memory");
#endif
}

// ---------------------------------------------------------------------------
// K0: zero the gram/rowsum accumulators (graph-replay safe re-init).
// ---------------------------------------------------------------------------
__global__ void zero_kernel(float* __restrict__ p, int n) {
    int i = blockIdx.x * 256 + threadIdx.x;
    if (i < n) p[i] = 0.f;
}

// ---------------------------------------------------------------------------
// K1: split-K partial Gram. grid (GROUPS, SPLIT), 128 threads (4 waves).
// 4-deep async global->LDS pipeline; WMMA 16x16x4 accumulation;
// fp32 atomic reduction of partial Gram + row sums into workspace.
// ---------------------------------------------------------------------------
__global__ __launch_bounds__(128) void gram_partial_kernel(
        const float* __restrict__ W,
        float* __restrict__ gramAcc,   // [GROUPS][32][32]
        float* __restrict__ sumAcc)    // [GROUPS*32]
{
    __shared__ float zb[NBUF][C][ZPAD];
    __shared__ float Sm[C][SPAD];
    __shared__ float rowsum[C];

    const int tid  = threadIdx.x;
    const int lane = tid & 31;
    const int wid  = tid >> 5;
    const int g    = blockIdx.x;
    const int ra0  = (wid >> 1) * 16;
    const int rb0  = (wid & 1) * 16;
    const int r    = lane & 15;
    const int hi   = (lane >= 16) ? 2 : 0;

    const float* Zg   = W + (size_t)g * C * DDIM;
    const int    kbeg = blockIdx.y * (DDIM / SPLIT);

    v8f acc = {};
    float rs = 0.f;

    // prologue: fill NBUF-1 chunks
#pragma unroll
    for (int p = 0; p < NBUF - 1; ++p)
        issue_chunk_async(Zg, kbeg + p * KC, zb[p], tid);

    for (int ci = 0; ci < NCHUNK; ++ci) {
        if (ci + NBUF - 1 < NCHUNK) {
            // safe: zb[(ci+3)&3] was last read at iter ci-1 (trailing barrier)
            issue_chunk_async(Zg, kbeg + (ci + NBUF - 1) * KC,
                              zb[(ci + NBUF - 1) & (NBUF - 1)], tid);
            wait_async_le12();            // own chunk-ci copies complete
        } else {
            wait_async_0();               // drain (no-op after first time)
        }
        __syncthreads();                  // all waves' chunk-ci data in LDS

        const float (*zt)[ZPAD] = zb[ci & (NBUF - 1)];
        if (wid == 0) {
#pragma unroll 8
            for (int c2 = 0; c2 < KC; ++c2) rs += zt[lane][c2];
        }
#pragma unroll 4
        for (int kk = 0; kk < KC; kk += 4) {
            v2f a = *(const v2f*)&zt[ra0 + r][kk + hi];
            v2f b = *(const v2f*)&zt[rb0 + r][kk + hi];
            acc = wmma4(a, b, acc);
        }
        __syncthreads();                  // done reading zt
    }

    if (wid == 0) rowsum[lane] = rs;
    store_tile(Sm, acc, ra0, rb0, lane);
    __syncthreads();

    for (int idx = tid; idx < C * C; idx += 128)
        unsafeAtomicAdd(&gramAcc[g * (C * C) + idx], Sm[idx >> 5][idx & 31]);
    if (tid < C)
        unsafeAtomicAdd(&sumAcc[g * C + tid], rowsum[tid]);
}

// ---------------------------------------------------------------------------
// K2: finalize S = (Gram - s s^T/d + eps I)/||.||_F, then 5 Newton-Schulz
// iterations with WMMA on LDS-resident 32x32 matrices. 64 blocks.
// ---------------------------------------------------------------------------
__global__ __launch_bounds__(128) void ns_kernel(
        const float* __restrict__ gramAcc,
        const float* __restrict__ sumAcc,
        float* __restrict__ Ball,
        float* __restrict__ meanOut,
        float* __restrict__ scaleOut)
{
    __shared__ float Sm[C][SPAD];
    __shared__ float Bm[C][SPAD];
    __shared__ float T1[C][SPAD];
    __shared__ float T2[C][SPAD];
    __shared__ float rowsum[C];
    __shared__ float red[4];
    __shared__ float invnorm_s, scale_s;

    const int tid  = threadIdx.x;
    const int lane = tid & 31;
    const int wid  = tid >> 5;
    const int g    = blockIdx.x;
    const int ra0  = (wid >> 1) * 16;
    const int rb0  = (wid & 1) * 16;
    const int r    = lane & 15;
    const int mo   = (lane >= 16) ? 8 : 0;

    if (tid < C) rowsum[tid] = sumAcc[g * C + tid];
    for (int idx = tid; idx < C * C; idx += 128)
        Sm[idx >> 5][idx & 31] = gramAcc[g * (C * C) + idx];
    __syncthreads();

    float ssq = 0.f;
    for (int idx = tid; idx < C * C; idx += 128) {
        int i = idx >> 5, j = idx & 31;
        float s = Sm[i][j] - rowsum[i] * rowsum[j] * (1.0f / (float)DDIM);
        if (i == j) s += EPS_VAL;
        Sm[i][j] = s;
        ssq += s * s;
    }
#pragma unroll
    for (int off = 16; off > 0; off >>= 1) ssq += __shfl_down(ssq, off, 32);
    if (lane == 0) red[wid] = ssq;
    __syncthreads();
    if (tid == 0) {
        float nrm = sqrtf(red[0] + red[1] + red[2] + red[3]);
        invnorm_s = 1.0f / nrm;
        scale_s   = rsqrtf(nrm);
    }
    __syncthreads();
    float inv = invnorm_s;
    for (int idx = tid; idx < C * C; idx += 128) {
        int i = idx >> 5, j = idx & 31;
        Sm[i][j] *= inv;
        Bm[i][j] = (i == j) ? 1.0f : 0.0f;
    }
    __syncthreads();

    for (int t = 0; t < T_ITERS; ++t) {
        v8f m = mm_tile(Bm, Bm, ra0, rb0, lane);       // B^2
        store_tile(T1, m, ra0, rb0, lane);
        __syncthreads();
        m = mm_tile(T1, Bm, ra0, rb0, lane);           // B^3
        store_tile(T2, m, ra0, rb0, lane);
        __syncthreads();
        m = mm_tile(T2, Sm, ra0, rb0, lane);           // B^3 @ S
#pragma unroll
        for (int v = 0; v < 8; ++v) {
            int row = ra0 + v + mo, col = rb0 + r;
            Bm[row][col] = 1.5f * Bm[row][col] - 0.5f * m[v];
        }
        __syncthreads();
    }

    for (int idx = tid; idx < C * C; idx += 128)
        Ball[g * (C * C) + idx] = Bm[idx >> 5][idx & 31];
    if (tid < C) meanOut[g * C + tid] = rowsum[tid] * (1.0f / (float)DDIM);
    if (tid == 0) scaleOut[g] = scale_s;
}

// ---------------------------------------------------------------------------
// K3: W = (B @ Zc) * scale. grid (GROUPS, DDIM/32), 128 threads.
// ---------------------------------------------------------------------------
__global__ __launch_bounds__(128) void ortho_apply_kernel(
        const float* __restrict__ W,
        const float* __restrict__ Ball,
        const float* __restrict__ meanB,
        const float* __restrict__ scaleB,
        float* __restrict__ out)
{
    __shared__ float Bl[C][SPAD];
    __shared__ float Zl[C][SPAD];

    const int tid  = threadIdx.x;
    const int lane = tid & 31;
    const int wid  = tid >> 5;
    const int g    = blockIdx.x;
    const int cb   = blockIdx.y * 32;
    const int ra0  = (wid >> 1) * 16;
    const int rb0  = (wid & 1) * 16;
    const int r    = lane & 15;
    const int mo   = (lane >= 16) ? 8 : 0;

    for (int idx = tid; idx < C * C; idx += 128) {
        int row = idx >> 5, col = idx & 31;
        Bl[row][col] = Ball[g * (C * C) + idx];
        Zl[row][col] = W[(size_t)(g * C + row) * DDIM + cb + col]
                       - meanB[g * C + row];
    }
    __syncthreads();

    v8f acc = mm_tile(Bl, Zl, ra0, rb0, lane);
    float sc = scaleB[g];
#pragma unroll
    for (int v = 0; v < 8; ++v) {
        int row = g * C + ra0 + v + mo;
        int col = cb + rb0 + r;
        out[(size_t)row * DDIM + col] = acc[v] * sc;
    }
}

extern "C" void kernel_launch(void* const* d_in, const int* in_sizes, int n_in,
                              void* d_out, int out_size, void* d_ws, size_t ws_size,
                              hipStream_t stream) {
    (void)in_sizes; (void)n_in; (void)out_size; (void)ws_size;
    const float* weight = (const float*)d_in[0];
    float* out = (float*)d_out;

    float* wsf     = (float*)d_ws;
    float* gramAcc = wsf;                          // 64*1024
    float* sumAcc  = gramAcc + GROUPS * C * C;     // 2048  (contiguous w/ gramAcc)
    float* Ball    = sumAcc + GROUPS * C;          // 64*1024
    float* meanB   = Ball + GROUPS * C * C;        // 2048
    float* scaleB  = meanB + GROUPS * C;           // 64

    const int nz = GROUPS * C * C + GROUPS * C;    // gramAcc + sumAcc
    zero_kernel<<<(nz + 255) / 256, 256, 0, stream>>>(gramAcc, nz);

    gram_partial_kernel<<<dim3(GROUPS, SPLIT), 128, 0, stream>>>(
        weight, gramAcc, sumAcc);
    ns_kernel<<<GROUPS, 128, 0, stream>>>(gramAcc, sumAcc, Ball, meanB, scaleB);
    ortho_apply_kernel<<<dim3(GROUPS, DDIM / 32), 128, 0, stream>>>(
        weight, Ball, meanB, scaleB, out);
}